// AFLB_9904194585088
// MI455X (gfx1250) — compile-verified
//
#include <hip/hip_runtime.h>
#include <math.h>

// ---------------- constants ----------------
constexpr int NB  = 4;
constexpr int CD  = 64;       // dim
constexpr int HH  = 192;
constexpr int WW  = 192;
constexpr int HWc = HH * WW;  // 36864

typedef float v2f __attribute__((ext_vector_type(2)));
typedef float v8f __attribute__((ext_vector_type(8)));

// =====================================================================
// Generic WMMA f32 GEMM:  O[b] = A[b](M x 64) @ X[b](64 x HWc) (+bias)
//   A batch stride aS (0 -> shared weights), X/O batch strides in elems.
//   Per wave: 16(M) x 64(P) tile via 4 accumulators, K-loop of 16 x (K=4).
// =====================================================================
__global__ void gemm_wmma(const float* __restrict__ A, int aS,
                          const float* __restrict__ X, long long xS,
                          float* __restrict__ O, long long oS,
                          const float* __restrict__ bias, int M, int Kd) {
  const int P = HWc;
  const int b = blockIdx.z;
  const float* Ab = A + (long long)b * aS;
  const float* Xb = X + (long long)b * xS;
  float* Ob = O + (long long)b * oS;
  const int lane = threadIdx.x & 31;
  const int wave = threadIdx.x >> 5;
  const int m0 = blockIdx.y * 64 + wave * 16;
  const int p0 = blockIdx.x * 64;
  const int mr = lane & 15;
  const int kh = (lane >> 4) << 1;  // 0 or 2 (K-half select, A/B layout)
  v8f acc0 = {}; v8f acc1 = {}; v8f acc2 = {}; v8f acc3 = {};
  for (int kb = 0; kb < Kd; kb += 4) {
    v2f a;
    a.x = Ab[(m0 + mr) * Kd + kb + kh];
    a.y = Ab[(m0 + mr) * Kd + kb + kh + 1];
    const float* xr0 = Xb + (long long)(kb + kh) * P;
    const float* xr1 = Xb + (long long)(kb + kh + 1) * P;
    v2f b0; b0.x = xr0[p0 + mr];      b0.y = xr1[p0 + mr];
    v2f b1; b1.x = xr0[p0 + 16 + mr]; b1.y = xr1[p0 + 16 + mr];
    v2f b2; b2.x = xr0[p0 + 32 + mr]; b2.y = xr1[p0 + 32 + mr];
    v2f b3; b3.x = xr0[p0 + 48 + mr]; b3.y = xr1[p0 + 48 + mr];
    acc0 = __builtin_amdgcn_wmma_f32_16x16x4_f32(false, a, false, b0, (short)0, acc0, false, false);
    acc1 = __builtin_amdgcn_wmma_f32_16x16x4_f32(false, a, false, b1, (short)0, acc1, false, false);
    acc2 = __builtin_amdgcn_wmma_f32_16x16x4_f32(false, a, false, b2, (short)0, acc2, false, false);
    acc3 = __builtin_amdgcn_wmma_f32_16x16x4_f32(false, a, false, b3, (short)0, acc3, false, false);
  }
  const int mhi = (lane >> 4) * 8;  // C/D layout: VGPR j -> M=j (lo half) / 8+j (hi half)
#pragma unroll
  for (int j = 0; j < 8; ++j) {
    int m = m0 + mhi + j;
    float bs = bias ? bias[m] : 0.f;
    Ob[(long long)m * P + p0 + mr]      = acc0[j] + bs;
    Ob[(long long)m * P + p0 + 16 + mr] = acc1[j] + bs;
    Ob[(long long)m * P + p0 + 32 + mr] = acc2[j] + bs;
    Ob[(long long)m * P + p0 + 48 + mr] = acc3[j] + bs;
  }
}

// ---------------- bilinear resize 96x96 -> 192x192 (half-pixel centers) ----
__global__ void k_resize(const float* __restrict__ x, float* __restrict__ xr) {
  int idx = blockIdx.x * 256 + threadIdx.x;
  if (idx >= NB * 3 * HWc) return;
  int p = idx % HWc, bc = idx / HWc;
  int yq = p / WW, xq = p % WW;
  float sy = 0.5f * yq - 0.25f, sx = 0.5f * xq - 0.25f;
  int y0 = (int)floorf(sy), x0 = (int)floorf(sx);
  float fy = sy - y0, fx = sx - x0;
  int y0c = min(max(y0, 0), 95), y1c = min(max(y0 + 1, 0), 95);
  int x0c = min(max(x0, 0), 95), x1c = min(max(x0 + 1, 0), 95);
  const float* ip = x + (long long)bc * 9216;
  float v = (1.f - fy) * ((1.f - fx) * ip[y0c * 96 + x0c] + fx * ip[y0c * 96 + x1c]) +
            fy * ((1.f - fx) * ip[y1c * 96 + x0c] + fx * ip[y1c * 96 + x1c]);
  xr[idx] = v;
}

// ---------------- conv1: 3x3, pad1, 3->64 -------------------------------
__global__ void k_conv1(const float* __restrict__ xr, const float* __restrict__ w,
                        float* __restrict__ out) {
  int idx = blockIdx.x * 256 + threadIdx.x;
  if (idx >= NB * CD * HWc) return;
  int p = idx % HWc, bo = idx / HWc;
  int oc = bo % CD, b = bo / CD;
  int y = p / WW, x = p % WW;
  float acc = 0.f;
  for (int ic = 0; ic < 3; ++ic) {
    const float* ip = xr + (long long)(b * 3 + ic) * HWc;
    const float* wp = w + (oc * 3 + ic) * 9;
#pragma unroll
    for (int ky = 0; ky < 3; ++ky) {
      int yy = y + ky - 1; if (yy < 0 || yy >= HH) continue;
#pragma unroll
      for (int kx = 0; kx < 3; ++kx) {
        int xx = x + kx - 1; if (xx < 0 || xx >= WW) continue;
        acc += ip[yy * WW + xx] * wp[ky * 3 + kx];
      }
    }
  }
  out[idx] = acc;
}

// ---------------- generic per-(b,c) plane reduce -------------------------
// mode 0/2: mean; mode 1: max(sqrt(sum sq),1e-12); mode 3: max
__global__ void k_row_reduce(const float* __restrict__ src, float* __restrict__ out,
                             int rowsPerBatch, int rowOff, int mode) {
  int c = blockIdx.x, b = blockIdx.y;
  const float* p = src + ((long long)b * rowsPerBatch + rowOff + c) * (long long)HWc;
  float acc = (mode == 3) ? -3.4e38f : 0.f;
  for (int i = threadIdx.x; i < HWc; i += 256) {
    float v = p[i];
    if (mode == 1) acc += v * v;
    else if (mode == 3) acc = fmaxf(acc, v);
    else acc += v;
  }
  __shared__ float red[256];
  red[threadIdx.x] = acc; __syncthreads();
  for (int s = 128; s > 0; s >>= 1) {
    if (threadIdx.x < s)
      red[threadIdx.x] = (mode == 3) ? fmaxf(red[threadIdx.x], red[threadIdx.x + s])
                                     : red[threadIdx.x] + red[threadIdx.x + s];
    __syncthreads();
  }
  if (threadIdx.x == 0) {
    float r = red[0];
    if (mode == 0 || mode == 2) r /= (float)HWc;
    if (mode == 1) { r = sqrtf(r); if (r < 1e-12f) r = 1e-12f; }
    out[b * CD + c] = r;
  }
}

// ---------------- rate MLP + circulant filter kernels --------------------
// gk layout per batch (768 floats): [gr_h | gi_h | gr_w | gi_w], each 192.
__global__ void k_rate_kernels(const float* __restrict__ meanbc,
                               const float* __restrict__ w1,
                               const float* __restrict__ w2,
                               float* __restrict__ gk) {
  int b = blockIdx.x, t = threadIdx.x;  // 192 threads
  __shared__ float tt[8];
  __shared__ int hs[2];
  if (t < 8) {
    float s = 0.f;
    for (int c = 0; c < CD; ++c) s += w1[t * CD + c] * meanbc[b * CD + c];
    tt[t] = 0.5f * s * (1.f + erff(s * 0.70710678f));  // exact GELU
  }
  __syncthreads();
  if (t < 2) {
    float s = 0.f;
    for (int o = 0; o < 8; ++o) s += w2[t * 8 + o] * tt[o];
    float thr = 1.f / (1.f + expf(-s));
    int v = (int)floorf(192.f * thr * 0.5f);
    hs[t] = v < 1 ? 1 : v;
  }
  __syncthreads();
  int h_ = hs[0], w_ = hs[1];
  float grh = 0.f, gih = 0.f, grw = 0.f, giw = 0.f;
  for (int k = 0; k < 192; ++k) {
    int s = (k + 96) % 192;
    float ang = 6.28318530718f * (float)((k * t) % 192) / 192.f;
    float ck = cosf(ang), sk = sinf(ang);
    if (s >= 96 - h_ && s < 96 + h_) { grh += ck; gih += sk; }
    if (s >= 96 - w_ && s < 96 + w_) { grw += ck; giw += sk; }
  }
  const float inv = 1.f / 192.f;
  gk[b * 768 + t]       = grh * inv;
  gk[b * 768 + 192 + t] = gih * inv;
  gk[b * 768 + 384 + t] = grw * inv;
  gk[b * 768 + 576 + t] = giw * inv;
}

// ---------------- frequency split: width pass (circular conv) ------------
__global__ void k_fft_width(const float* __restrict__ x1, const float* __restrict__ gk,
                            float* __restrict__ tr, float* __restrict__ ti) {
  int gid = blockIdx.x;               // b*64*192 + c*192 + r
  int r = gid % HH, bc = gid / HH, b = bc / CD;
  const float* row = x1 + (long long)bc * HWc + r * WW;
  __shared__ float srow[192], sgr[192], sgi[192];
  int t = threadIdx.x;  // 192
  srow[t] = row[t];
  sgr[t] = gk[b * 768 + 384 + t];
  sgi[t] = gk[b * 768 + 576 + t];
  __syncthreads();
  float ar = 0.f, ai = 0.f;
  for (int j = 0; j < 192; ++j) {
    int d = t - j; d += (d < 0) ? 192 : 0;
    float v = srow[j];
    ar += v * sgr[d];
    ai += v * sgi[d];
  }
  long long o = (long long)bc * HWc + r * WW + t;
  tr[o] = ar; ti[o] = ai;
}

// ---------------- height pass + combine: low, high = x1 - low ------------
__global__ void k_fft_height(const float* __restrict__ x1, const float* __restrict__ tr,
                             const float* __restrict__ ti, const float* __restrict__ gk,
                             float* __restrict__ low, float* __restrict__ high) {
  int bc = blockIdx.x, b = bc / CD;
  int r0 = blockIdx.y * 32;
  int x = threadIdx.x;  // 192
  __shared__ float sgr[192], sgi[192];
  sgr[x] = gk[b * 768 + x];
  sgi[x] = gk[b * 768 + 192 + x];
  __syncthreads();
  float acc[32];
#pragma unroll
  for (int i = 0; i < 32; ++i) acc[i] = 0.f;
  const float* trp = tr + (long long)bc * HWc;
  const float* tip = ti + (long long)bc * HWc;
  for (int rp = 0; rp < 192; ++rp) {
    float vr = trp[rp * WW + x];
    float vi = tip[rp * WW + x];
#pragma unroll
    for (int i = 0; i < 32; ++i) {
      int d = r0 + i - rp; d += (d < 0) ? 192 : 0;
      acc[i] += vr * sgr[d] - vi * sgi[d];
    }
  }
#pragma unroll
  for (int i = 0; i < 32; ++i) {
    long long idx = (long long)bc * HWc + (r0 + i) * WW + x;
    float lo = acc[i];
    low[idx] = lo;
    high[idx] = x1[idx] - lo;
  }
}

// ---------------- depthwise 3x3, pad1 ------------------------------------
__global__ void k_dw3x3(const float* __restrict__ in, const float* __restrict__ w,
                        float* __restrict__ out, int Cch) {
  long long idx = (long long)blockIdx.x * 256 + threadIdx.x;
  if (idx >= (long long)NB * Cch * HWc) return;
  int p = (int)(idx % HWc);
  long long bc = idx / HWc;
  int ch = (int)(bc % Cch);
  int y = p / WW, x = p % WW;
  const float* ip = in + bc * HWc;
  const float* wp = w + ch * 9;
  float acc = 0.f;
#pragma unroll
  for (int ky = 0; ky < 3; ++ky) {
    int yy = y + ky - 1; if (yy < 0 || yy >= HH) continue;
#pragma unroll
    for (int kx = 0; kx < 3; ++kx) {
      int xx = x + kx - 1; if (xx < 0 || xx >= WW) continue;
      acc += ip[yy * WW + xx] * wp[ky * 3 + kx];
    }
  }
  out[idx] = acc;
}

// ---------------- zero fill ---------------------------------------------
__global__ void k_zero(float* __restrict__ p, int n) {
  int i = blockIdx.x * 256 + threadIdx.x;
  if (i < n) p[i] = 0.f;
}

// ---------------- q@k^T split-K GEMM with atomic accumulation ------------
__global__ void k_qk_gemm(const float* __restrict__ q, const float* __restrict__ kv,
                          float* __restrict__ attnraw) {
  const int b = blockIdx.y;
  const int CHUNK = HWc / 16;  // 2304
  const int pbeg = blockIdx.x * CHUNK;
  __shared__ float qs[64][65];
  __shared__ float ks[64][65];
  int tid = threadIdx.x;            // 256
  int trw = tid >> 4, tcl = tid & 15;
  float acc[4][4];
#pragma unroll
  for (int i = 0; i < 4; ++i)
#pragma unroll
    for (int j = 0; j < 4; ++j) acc[i][j] = 0.f;
  const float* qb = q + (long long)b * CD * HWc;
  const float* kb = kv + (long long)b * 2 * CD * HWc;  // k = first 64 rows
  for (int p0 = pbeg; p0 < pbeg + CHUNK; p0 += 64) {
    for (int i = tid; i < 4096; i += 256) {
      int c = i >> 6, pl = i & 63;
      qs[c][pl] = qb[(long long)c * HWc + p0 + pl];
      ks[c][pl] = kb[(long long)c * HWc + p0 + pl];
    }
    __syncthreads();
    for (int kk = 0; kk < 64; ++kk) {
      float a0 = qs[trw * 4 + 0][kk], a1 = qs[trw * 4 + 1][kk];
      float a2 = qs[trw * 4 + 2][kk], a3 = qs[trw * 4 + 3][kk];
      float b0 = ks[tcl * 4 + 0][kk], b1 = ks[tcl * 4 + 1][kk];
      float b2 = ks[tcl * 4 + 2][kk], b3 = ks[tcl * 4 + 3][kk];
      acc[0][0] += a0 * b0; acc[0][1] += a0 * b1; acc[0][2] += a0 * b2; acc[0][3] += a0 * b3;
      acc[1][0] += a1 * b0; acc[1][1] += a1 * b1; acc[1][2] += a1 * b2; acc[1][3] += a1 * b3;
      acc[2][0] += a2 * b0; acc[2][1] += a2 * b1; acc[2][2] += a2 * b2; acc[2][3] += a2 * b3;
      acc[3][0] += a3 * b0; acc[3][1] += a3 * b1; acc[3][2] += a3 * b2; acc[3][3] += a3 * b3;
    }
    __syncthreads();
  }
#pragma unroll
  for (int i = 0; i < 4; ++i)
#pragma unroll
    for (int j = 0; j < 4; ++j)
      atomicAdd(&attnraw[b * 4096 + (trw * 4 + i) * 64 + (tcl * 4 + j)], acc[i][j]);
}

// ---------------- fold norms + temp, softmax over d ----------------------
__global__ void k_attn_softmax(const float* __restrict__ raw, const float* __restrict__ nq,
                               const float* __restrict__ nk, const float* __restrict__ temp,
                               float* __restrict__ attn) {
  int c = blockIdx.x, b = blockIdx.y, d = threadIdx.x;  // 64 threads
  __shared__ float red[64];
  float v = raw[b * 4096 + c * 64 + d] / (nq[b * CD + c] * nk[b * CD + d]) * temp[0];
  red[d] = v; __syncthreads();
  for (int s = 32; s > 0; s >>= 1) {
    if (d < s) red[d] = fmaxf(red[d], red[d + s]);
    __syncthreads();
  }
  float mx = red[0]; __syncthreads();
  float e = expf(v - mx);
  red[d] = e; __syncthreads();
  for (int s = 32; s > 0; s >>= 1) {
    if (d < s) red[d] += red[d + s];
    __syncthreads();
  }
  attn[b * 4096 + c * 64 + d] = e / red[0];
}

// ---------------- spatial gate: channel max + mean -----------------------
__global__ void k_chanstats(const float* __restrict__ t, float* __restrict__ spin) {
  int idx = blockIdx.x * 256 + threadIdx.x;
  if (idx >= NB * HWc) return;
  int b = idx / HWc, p = idx % HWc;
  float mx = -3.4e38f, sm = 0.f;
  for (int c = 0; c < CD; ++c) {
    float v = t[((long long)b * CD + c) * HWc + p];
    mx = fmaxf(mx, v); sm += v;
  }
  spin[(long long)b * 2 * HWc + p] = mx;
  spin[(long long)b * 2 * HWc + HWc + p] = sm / (float)CD;
}

// ---------------- 7x7 conv (2->1), pad3, sigmoid -------------------------
__global__ void k_conv7(const float* __restrict__ spin, const float* __restrict__ w,
                        float* __restrict__ sw) {
  int idx = blockIdx.x * 256 + threadIdx.x;
  if (idx >= NB * HWc) return;
  int b = idx / HWc, p = idx % HWc;
  int y = p / WW, x = p % WW;
  float acc = 0.f;
  for (int ic = 0; ic < 2; ++ic) {
    const float* ip = spin + (long long)(b * 2 + ic) * HWc;
    const float* wp = w + ic * 49;
    for (int ky = 0; ky < 7; ++ky) {
      int yy = y + ky - 3; if (yy < 0 || yy >= HH) continue;
      for (int kx = 0; kx < 7; ++kx) {
        int xx = x + kx - 3; if (xx < 0 || xx >= WW) continue;
        acc += ip[yy * WW + xx] * wp[ky * 7 + kx];
      }
    }
  }
  sw[idx] = 1.f / (1.f + expf(-acc));
}

// ---------------- channel gate MLP ---------------------------------------
__global__ void k_chan_mlp(const float* __restrict__ avg, const float* __restrict__ mx,
                           const float* __restrict__ w1, const float* __restrict__ w2,
                           float* __restrict__ cw) {
  int b = blockIdx.x, c = threadIdx.x;  // 64 threads
  __shared__ float fa[64], fm[64], h1a[4], h1m[4];
  fa[c] = avg[b * CD + c];
  fm[c] = mx[b * CD + c];
  __syncthreads();
  if (c < 4) {
    float s = 0.f, s2 = 0.f;
    for (int i = 0; i < CD; ++i) { s += w1[c * CD + i] * fa[i]; s2 += w1[c * CD + i] * fm[i]; }
    h1a[c] = fmaxf(s, 0.f); h1m[c] = fmaxf(s2, 0.f);
  }
  __syncthreads();
  float o = 0.f;
  for (int j = 0; j < 4; ++j) o += w2[c * 4 + j] * (h1a[j] + h1m[j]);
  cw[b * CD + c] = 1.f / (1.f + expf(-o));
}

// ---------------- fre_refine combine -------------------------------------
__global__ void k_fre_combine(const float* __restrict__ lo2, const float* __restrict__ hi2,
                              const float* __restrict__ sw, const float* __restrict__ cw,
                              float* __restrict__ comb) {
  int idx = blockIdx.x * 256 + threadIdx.x;
  if (idx >= NB * CD * HWc) return;
  int p = idx % HWc;
  int bc = idx / HWc;
  int c = bc % CD, b = bc / CD;
  comb[idx] = lo2[idx] * sw[b * HWc + p] + hi2[idx] * cw[b * CD + c];
}

// ---------------- final: out = cca*para1 + y*para2 -----------------------
__global__ void k_final(const float* __restrict__ cc, const float* __restrict__ y,
                        const float* __restrict__ p1, const float* __restrict__ p2,
                        float* __restrict__ out) {
  int idx = blockIdx.x * 256 + threadIdx.x;
  if (idx >= NB * CD * HWc) return;
  int c = (idx / HWc) % CD;
  out[idx] = cc[idx] * p1[c] + y[idx] * p2[c];
}

// =====================================================================
// host-side orchestration
// =====================================================================
static inline void launch_gemm(const float* A, int aS, const float* X, long long xS,
                               float* O, long long oS, const float* bias, int M,
                               hipStream_t st) {
  dim3 g(HWc / 64, M / 64, NB);
  gemm_wmma<<<g, 128, 0, st>>>(A, aS, X, xS, O, oS, bias, M, 64);
}

static void run_cca(const float* xin, const float* yin,
                    const float* qw, const float* qdw, const float* kvw,
                    const float* kvdw, const float* projw, const float* temp,
                    float* sQpre, float* sQ, float* sKvp, float* sKv, float* sOutpre,
                    float* outFinal, float* attnraw, float* attn, float* nq, float* nk,
                    hipStream_t st) {
  const long long TS1 = (long long)CD * HWc;      // 64 x 36864
  const long long TS2 = 2LL * CD * HWc;           // 128 x 36864
  // q = dw3x3(conv1x1(x))
  launch_gemm(qw, 0, xin, TS1, sQpre, TS1, nullptr, 64, st);
  {
    long long n = (long long)NB * CD * HWc;
    k_dw3x3<<<(unsigned)((n + 255) / 256), 256, 0, st>>>(sQpre, qdw, sQ, CD);
  }
  // kv = dw3x3(conv1x1(y)), 128 channels
  launch_gemm(kvw, 0, yin, TS1, sKvp, TS2, nullptr, 128, st);
  {
    long long n = (long long)NB * 2 * CD * HWc;
    k_dw3x3<<<(unsigned)((n + 255) / 256), 256, 0, st>>>(sKvp, kvdw, sKv, 2 * CD);
  }
  // attn logits
  k_zero<<<(NB * 4096 + 255) / 256, 256, 0, st>>>(attnraw, NB * 4096);
  k_row_reduce<<<dim3(CD, NB), 256, 0, st>>>(sQ, nq, CD, 0, 1);
  k_row_reduce<<<dim3(CD, NB), 256, 0, st>>>(sKv, nk, 2 * CD, 0, 1);
  k_qk_gemm<<<dim3(16, NB), 256, 0, st>>>(sQ, sKv, attnraw);
  k_attn_softmax<<<dim3(CD, NB), 64, 0, st>>>(attnraw, nq, nk, temp, attn);
  // out = attn @ v  (v = second half of kv)
  launch_gemm(attn, 4096, sKv + (long long)CD * HWc, TS2, sOutpre, TS1, nullptr, 64, st);
  // proj
  launch_gemm(projw, 0, sOutpre, TS1, outFinal, TS1, nullptr, 64, st);
}

extern "C" void kernel_launch(void* const* d_in, const int* in_sizes, int n_in,
                              void* d_out, int out_size, void* d_ws, size_t ws_size,
                              hipStream_t stream) {
  (void)in_sizes; (void)n_in; (void)out_size; (void)ws_size;
  const float* x          = (const float*)d_in[0];
  const float* y          = (const float*)d_in[1];
  const float* conv1w     = (const float*)d_in[2];
  const float* rate_w1    = (const float*)d_in[3];
  const float* rate_w2    = (const float*)d_in[4];
  const float* para1      = (const float*)d_in[5];
  const float* para2      = (const float*)d_in[6];
  const float* fr_spatial = (const float*)d_in[7];
  const float* fr_ch_w1   = (const float*)d_in[8];
  const float* fr_ch_w2   = (const float*)d_in[9];
  const float* fr_proj_w  = (const float*)d_in[10];
  const float* fr_proj_b  = (const float*)d_in[11];
  // per-prefix params: temp, q_w, q_dw, kv_w, kv_dw, proj  (l=12.., h=18.., agg=24..)
  const float* P_[3][6];
  for (int g = 0; g < 3; ++g)
    for (int j = 0; j < 6; ++j) P_[g][j] = (const float*)d_in[12 + g * 6 + j];

  float* ws = (float*)d_ws;
  const size_t TS = (size_t)NB * CD * HWc;  // one (4,64,192,192) slot
  float* sA    = ws + 0 * TS;   // x1, later q_pre / comb
  float* sB    = ws + 1 * TS;   // tmp_r, later q
  float* sC    = ws + 2 * TS;   // tmp_i, later out_pre
  float* sLow  = ws + 3 * TS;   // low, later agg
  float* sHigh = ws + 4 * TS;   // high, later cca('agg') output
  float* sKvp  = ws + 5 * TS;   // 2 slots (5,6)
  float* sKv   = ws + 7 * TS;   // 2 slots (7,8)
  float* sHi2  = ws + 9 * TS;   // cca 'l' output
  float* sLo2  = ws + 10 * TS;  // cca 'h' output
  float* sm    = ws + 11 * TS;  // small buffers
  float* meanbc = sm + 0;        // 256
  float* gk     = sm + 256;      // 3072
  float* nq     = sm + 3328;     // 256
  float* nk     = sm + 3584;     // 256
  float* attnr  = sm + 3840;     // 16384
  float* attn   = sm + 20224;    // 16384
  float* spin   = sm + 36608;    // 294912
  float* sw     = sm + 331520;   // 147456
  float* cgavg  = sm + 478976;   // 256
  float* cgmax  = sm + 479232;   // 256
  float* cw     = sm + 479488;   // 256
  float* xr     = sm + 479744;   // 442368

  const int NFULL = NB * CD * HWc;  // 9,437,184

  // 1) bilinear resize x -> 192x192
  k_resize<<<(NB * 3 * HWc + 255) / 256, 256, 0, stream>>>(x, xr);
  // 2) conv1 3x3 -> x1 (sA)
  k_conv1<<<(NFULL + 255) / 256, 256, 0, stream>>>(xr, conv1w, sA);
  // 3) adaptive avg pool + rate MLP + circulant kernels
  k_row_reduce<<<dim3(CD, NB), 256, 0, stream>>>(sA, meanbc, CD, 0, 0);
  k_rate_kernels<<<NB, 192, 0, stream>>>(meanbc, rate_w1, rate_w2, gk);
  // 4) frequency split (separable circular convolution == masked FFT)
  k_fft_width<<<NB * CD * HH, 192, 0, stream>>>(sA, gk, sB, sC);
  k_fft_height<<<dim3(NB * CD, HH / 32), 192, 0, stream>>>(sA, sB, sC, gk, sLow, sHigh);

  // 5) cca('l', high, y) -> sHi2 ; cca('h', low, y) -> sLo2
  run_cca(sHigh, y, P_[0][1], P_[0][2], P_[0][3], P_[0][4], P_[0][5], P_[0][0],
          sA, sB, sKvp, sKv, sC, sHi2, attnr, attn, nq, nk, stream);
  run_cca(sLow, y, P_[1][1], P_[1][2], P_[1][3], P_[1][4], P_[1][5], P_[1][0],
          sA, sB, sKvp, sKv, sC, sLo2, attnr, attn, nq, nk, stream);

  // 6) fre_refine(low=sLo2, high=sHi2) -> agg (sLow)
  k_chanstats<<<(NB * HWc + 255) / 256, 256, 0, stream>>>(sHi2, spin);
  k_conv7<<<(NB * HWc + 255) / 256, 256, 0, stream>>>(spin, fr_spatial, sw);
  k_row_reduce<<<dim3(CD, NB), 256, 0, stream>>>(sLo2, cgavg, CD, 0, 2);
  k_row_reduce<<<dim3(CD, NB), 256, 0, stream>>>(sLo2, cgmax, CD, 0, 3);
  k_chan_mlp<<<NB, 64, 0, stream>>>(cgavg, cgmax, fr_ch_w1, fr_ch_w2, cw);
  k_fre_combine<<<(NFULL + 255) / 256, 256, 0, stream>>>(sLo2, sHi2, sw, cw, sA);
  launch_gemm(fr_proj_w, 0, sA, (long long)CD * HWc, sLow, (long long)CD * HWc,
              fr_proj_b, 64, stream);

  // 7) cca('agg', y, agg) -> sHigh
  run_cca(y, sLow, P_[2][1], P_[2][2], P_[2][3], P_[2][4], P_[2][5], P_[2][0],
          sA, sB, sKvp, sKv, sC, sHigh, attnr, attn, nq, nk, stream);

  // 8) out = cca_out * para1 + y * para2
  k_final<<<(NFULL + 255) / 256, 256, 0, stream>>>(sHigh, y, para1, para2,
                                                   (float*)d_out);
}